// SegmentPager_5944234738185
// MI455X (gfx1250) — compile-verified
//
#include <hip/hip_runtime.h>
#include <stdint.h>

#define B  32
#define L  8192
#define P  64
#define MP 8192   // MAX_PAGES

// ---- output layout (floats, concatenated in reference return order) ----
static constexpr size_t OFF_TV   = 0;                         // token_valid  [B,L]
static constexpr size_t OFF_PIDX = (size_t)B * L;             // page_indices [B,MP,P]
static constexpr size_t N_PAGES  = (size_t)B * MP * P;        // 16,777,216
static constexpr size_t OFF_PVAL = OFF_PIDX + N_PAGES;        // page_valid   [B,MP,P]
static constexpr size_t OFF_SEG  = OFF_PVAL + N_PAGES;        // segment_ids  [B,MP]
static constexpr size_t OFF_T2P  = OFF_SEG + (size_t)B * MP;  // token2page   [B,L]

typedef unsigned int v4u __attribute__((ext_vector_type(4)));
typedef int          v4i __attribute__((ext_vector_type(4)));
typedef int          v8i __attribute__((ext_vector_type(8)));

// Hard-require the TDM builtin in the DEVICE pass only (host pass lacks
// AMDGCN builtins by design; its copy of device code is never codegen'd).
#if defined(__HIP_DEVICE_COMPILE__)
#if !__has_builtin(__builtin_amdgcn_tensor_load_to_lds)
#error "gfx1250 TDM builtin __builtin_amdgcn_tensor_load_to_lds not available on device"
#endif
#endif

// ---------------------------------------------------------------------------
// Kernel 1: default fill of the big tables (store-bandwidth bound, ~135 MB).
// page_indices := -1, page_valid := 0, segment_ids := -1. float4 stores
// (global_store_b128, verified in disasm).
// ---------------------------------------------------------------------------
__global__ __launch_bounds__(256) void fill_kernel(float* __restrict__ out) {
  const size_t n1     = N_PAGES;                     // page_indices count
  const size_t total  = 2 * N_PAGES + (size_t)B * MP;
  const size_t total4 = total >> 2;
  float4* o = (float4*)(out + OFF_PIDX);
  size_t idx    = (size_t)blockIdx.x * blockDim.x + threadIdx.x;
  size_t stride = (size_t)gridDim.x * blockDim.x;
  for (size_t v = idx; v < total4; v += stride) {
    size_t e = v << 2;
    float val = (e >= n1 && e < 2 * n1) ? 0.0f : -1.0f;
    o[v] = make_float4(val, val, val, val);
  }
}

// ---------------------------------------------------------------------------
// TDM helper: 1-row (8192 x int32) tile DMA from global to LDS.
// D# packed per CDNA5 ISA §8.3/8.4 (group0: count/lds/global/type,
// group1: data_size=4B, tensor 8192x1, tile 8192x1, stride0=8192).
// ---------------------------------------------------------------------------
__device__ __forceinline__ void tdm_load_row(const int* __restrict__ gsrc,
                                             int* __restrict__ ldst) {
#if __has_builtin(__builtin_amdgcn_tensor_load_to_lds)
  unsigned lds_off = (unsigned)(uintptr_t)(void*)ldst;
  unsigned long long ga = (unsigned long long)(uintptr_t)gsrc;
  v4u g0 = { 1u,                                         // count=1
             lds_off,                                    // lds_addr
             (unsigned)(ga & 0xffffffffull),             // global_addr lo32
             (unsigned)((ga >> 32) & 0x01ffffffull) | (2u << 30) }; // hi | type=2
  v8i g1 = { (int)0x00020000u,                        // data_size=2 (4 bytes)
             (int)((unsigned)L << 16),                // tensor_dim0[15:0]
             (int)(((unsigned)L >> 16) | (1u << 16)), // dim0 hi | tensor_dim1=1
             (int)((unsigned)L << 16),                // dim1 hi | tile_dim0=8192
             1,                                       // tile_dim1=1, tile_dim2=0
             (int)L,                                  // tensor_dim0_stride lo32
             (int)((unsigned)L << 16),                // stride0 hi | stride1 lo16
             0 };                                     // stride1 hi
  v4i z4 = { 0, 0, 0, 0 };
#if __clang_major__ >= 23
  v8i z8 = { 0, 0, 0, 0, 0, 0, 0, 0 };
  __builtin_amdgcn_tensor_load_to_lds(g0, g1, z4, z4, z8, 0);
#else
  __builtin_amdgcn_tensor_load_to_lds(g0, g1, z4, z4, 0);
#endif
#else
  // Host-pass stub (never executed; device pass is guaranteed the builtin
  // by the #error above). Cooperative wave-0 copy for semantic completeness.
  for (int i = (int)(threadIdx.x & 31); i < L; i += 32) ldst[i] = gsrc[i];
#endif
}

__device__ __forceinline__ void tdm_wait_all() {
#if __has_builtin(__builtin_amdgcn_s_wait_tensorcnt)
  __builtin_amdgcn_s_wait_tensorcnt(0);
#endif
}

// ---------------------------------------------------------------------------
// Kernel 2: one block per row. TDM-stage tokens+mask into LDS, build flags,
// block scan, tiny sequential per-segment scan (~17 segments expected),
// parallel coalesced scatter.
// ---------------------------------------------------------------------------
__global__ __launch_bounds__(256) void pager_kernel(
    const int* __restrict__ input_ids,
    const int* __restrict__ attention_mask,
    const int* __restrict__ query_pos,
    float* __restrict__ out)
{
  __shared__ int            s_tok[L];          // 32 KB (TDM destination)
  __shared__ int            s_msk[L];          // 32 KB (TDM destination)
  __shared__ unsigned       s_flagbits[L/32];  // 1 KB
  __shared__ unsigned       s_validbits[L/32]; // 1 KB
  __shared__ short          s_starts[L];       // 16 KB (start pos per segment)
  __shared__ unsigned char  s_cstart[L];       // 8 KB  (page fill at seg start)
  __shared__ unsigned short s_firstpage[L];    // 16 KB (page id of seg's 1st token)
  __shared__ int            s_scan[256];
  __shared__ int            s_any;

  const int b   = blockIdx.x;
  const int tid = threadIdx.x;
  int qp = query_pos[b];
  qp = qp < 0 ? 0 : (qp > L ? L : qp);

  // ---- Phase 0: stage token + mask rows into LDS via Tensor Data Mover ----
  if (tid < 32) {  // wave 0 issues both DMAs (operands are wave-uniform SGPRs)
    tdm_load_row(input_ids      + (size_t)b * L, s_tok);
    tdm_load_row(attention_mask + (size_t)b * L, s_msk);
    tdm_wait_all();
  }
  if (tid == 0) s_any = 0;
  __syncthreads();

  // ---- Phase 1: per-thread 32-token chunk: start flags + valid bits ----
  const int base = tid << 5;
  unsigned fbits = 0, vbits = 0;
  int cnt = 0;
  int prev = (tid == 0) ? -1 : s_tok[base - 1];
  for (int j = 0; j < 32; ++j) {
    int i = base + j;
    bool st = (i == 0) | (prev == 13) | (i == qp);
    if (st) { fbits |= 1u << j; ++cnt; }
    if (s_msk[i]) vbits |= 1u << j;
    prev = s_tok[i];
  }
  s_flagbits[tid]  = fbits;
  s_validbits[tid] = vbits;
  s_scan[tid]      = cnt;
  __syncthreads();
  if (vbits) atomicOr(&s_any, 1);

  // ---- Phase 2: block inclusive scan over per-thread start counts ----
  for (int off = 1; off < 256; off <<= 1) {
    int add = (tid >= off) ? s_scan[tid - off] : 0;
    __syncthreads();
    s_scan[tid] += add;
    __syncthreads();
  }
  const int excl    = s_scan[tid] - cnt;
  const int nStarts = s_scan[255];

  // ---- Phase 3: compact start positions ----
  {
    int c = excl;
    for (int j = 0; j < 32; ++j)
      if (fbits & (1u << j)) s_starts[c++] = (short)(base + j);
  }
  __syncthreads();

  // ---- Phase 4: sequential per-segment scan (expected ~17 segments) ----
  if (tid == 0) {
    int cc = 0, p = -1;
    for (int k = 0; k < nStarts; ++k) {
      int st  = s_starts[k];
      int nxt = (k + 1 < nStarts) ? (int)s_starts[k + 1] : L;
      int len = nxt - st;
      bool qs = (st == qp);
      if ((qs && cc > 0) || (len > P - cc)) cc = 0;   // flush
      s_cstart[k] = (unsigned char)cc;
      if (cc == 0) ++p;                               // first token opens a page
      s_firstpage[k] = (unsigned short)p;
      p = p + (cc + len - 1) / P;                     // page of last token
      cc = (cc + len) & (P - 1);
    }
  }
  __syncthreads();

  // ---- Phase 5: coalesced parallel scatter ----
  float* tv      = out + OFF_TV   + (size_t)b * L;
  float* pidx    = out + OFF_PIDX + (size_t)b * MP * P;
  float* pval    = out + OFF_PVAL + (size_t)b * MP * P;
  float* seg_out = out + OFF_SEG  + (size_t)b * MP;
  float* t2p     = out + OFF_T2P  + (size_t)b * L;
  const bool anyv = (s_any != 0);

  for (int i = tid; i < L; i += 256) {
    bool v = (s_validbits[i >> 5] >> (i & 31)) & 1u;
    tv[i] = v ? 1.0f : 0.0f;
    if (anyv) {
      // largest k with starts[k] <= i
      int lo = 0, hi = nStarts - 1;
      while (lo < hi) {
        int mid = (lo + hi + 1) >> 1;
        if ((int)s_starts[mid] <= i) lo = mid; else hi = mid - 1;
      }
      int k    = lo;
      int t    = (int)s_cstart[k] + (i - (int)s_starts[k]);
      int page = (int)s_firstpage[k] + (t >> 6);
      int slot = t & 63;
      int f    = (page << 6) + slot;
      t2p[i]  = (float)page;
      pidx[f] = (float)i;
      pval[f] = v ? 1.0f : 0.0f;
      if (slot == 0) seg_out[page] = (float)k;   // new_page <=> slot==0
    } else {
      t2p[i] = -1.0f;  // degenerate row (token_valid all false already)
    }
  }
}

extern "C" void kernel_launch(void* const* d_in, const int* in_sizes, int n_in,
                              void* d_out, int out_size, void* d_ws, size_t ws_size,
                              hipStream_t stream) {
  (void)in_sizes; (void)n_in; (void)out_size; (void)d_ws; (void)ws_size;
  const int* input_ids      = (const int*)d_in[0];
  const int* attention_mask = (const int*)d_in[1];
  const int* query_pos      = (const int*)d_in[2];
  float* out = (float*)d_out;

  fill_kernel<<<4096, 256, 0, stream>>>(out);
  pager_kernel<<<B, 256, 0, stream>>>(input_ids, attention_mask, query_pos, out);
}